// AnchorGenerator_89919435309502
// MI455X (gfx1250) — compile-verified
//
#include <hip/hip_runtime.h>

// AnchorGenerator for (1,256,512,512) feature map, BASE_SIZE=16,
// scales={0.5,1,2}, ratios={0.5,1,2}.
// Output: (512*512*9, 4) f32 anchors, flat = 9,437,184 floats = 37.75 MB.
//
// Pure store-roofline kernel: ~1.6 us at 23.3 TB/s. Zero matrix flops ->
// WMMA inapplicable. CDNA5-relevant paths: wave32-coalesced
// global_store_b128 with non-temporal (TH=NT) cache policy.
//
// Division-free: hh = (16*scale/sqrt_r)/2 == 2^(2+s) * round(1/sqrt_r),
// since rounding commutes with exact power-of-two scaling. The f32
// reciprocals of the f32 sqrt constants round exactly to the mirrored
// sqrt constants (verified): 1/f32(sqrt(.5)) -> 0x3FB504F3, and
// 1/f32(sqrt(2)) -> 0x3F3504F3. So hw/hh are cndmask-selects + v_ldexp.

#define H_DIM 512
#define W_DIM 512

// Native vector type: required by __builtin_nontemporal_store.
typedef __attribute__((ext_vector_type(4))) float v4f;

__global__ __launch_bounds__(256) void
AnchorGenerator_89919435309502_kernel(v4f* __restrict__ out, unsigned n) {
    unsigned i = blockIdx.x * blockDim.x + threadIdx.x;
    if (i >= n) return;

    // i = ((h*512 + w)*9 + s*3 + r)
    unsigned pos = i / 9u;          // h*512 + w  (magic-multiply)
    unsigned a   = i - pos * 9u;    // anchor id: s*3 + r
    unsigned w   = pos & (W_DIM - 1u);
    unsigned h   = pos >> 9;
    unsigned s   = a / 3u;          // scale index: 0,1,2 -> 0.5,1,2
    unsigned r   = a - s * 3u;      // ratio index: 0,1,2 -> 0.5,1,2

    const float SQH = 0.70710678118654752440f;  // f32(sqrt(0.5)) = 0x3F3504F3
    const float SQ2 = 1.41421356237309504880f;  // f32(sqrt(2))   = 0x3FB504F3

    // sr = sqrt(ratio); rsr = round(1/sqrt(ratio)) = mirrored constant.
    // Two v_cmp shared by four v_cndmask.
    bool r1 = (r == 1u), r2 = (r == 2u);
    float sr  = r1 ? 1.0f : (r2 ? SQ2 : SQH);
    float rsr = r1 ? 1.0f : (r2 ? SQH : SQ2);

    // hw = 8*scale*sr ; hh = 8*scale/sr ; 8*scale = 2^(2+s) -> exact ldexp.
    int e = (int)s + 2;
    float hw = ldexpf(sr,  e);
    float hh = ldexpf(rsr, e);

    float xc = fmaf((float)w, 16.0f, 8.0f);   // exact in f32
    float yc = fmaf((float)h, 16.0f, 8.0f);   // exact in f32
    const float xmax = (float)(W_DIM * 16);   // 8192
    const float ymax = (float)(H_DIM * 16);   // 8192

    v4f v;
    v.x = fmaxf(xc - hw, 0.0f);
    v.y = fmaxf(yc - hh, 0.0f);
    v.z = fminf(xc + hw, xmax);
    v.w = fminf(yc + hh, ymax);

    // Write-once streaming output: non-temporal 128-bit store (TH=NT),
    // lane-contiguous 16B chunks -> fully coalesced global_store_b128.
    __builtin_nontemporal_store(v, out + i);
}

extern "C" void kernel_launch(void* const* d_in, const int* in_sizes, int n_in,
                              void* d_out, int out_size, void* d_ws, size_t ws_size,
                              hipStream_t stream) {
    (void)d_in; (void)in_sizes; (void)n_in; (void)d_ws; (void)ws_size;
    // out_size = H*W*9*4 floats; one v4f per anchor.
    unsigned n = (unsigned)(out_size / 4);            // 2,359,296 anchors
    dim3 block(256);                                   // 8 wave32 per block
    dim3 grid((n + block.x - 1) / block.x);            // 9216 blocks exactly
    AnchorGenerator_89919435309502_kernel<<<grid, block, 0, stream>>>(
        (v4f*)d_out, n);
}